// FusedGroupedMLP_721554506098
// MI455X (gfx1250) — compile-verified
//
#include <hip/hip_runtime.h>

#define B_ 2
#define N_ 25
#define S_ 4096
#define I_ 63
#define W_ 256
#define D_ 8
#define Z_ 16

#define TS    128   // tokens per workgroup (one 16-token m-tile per wave, 8 waves)
#define HSTR  260   // h_buf row stride in floats: 260 mod 64 == 4 -> conflict-free b64 reads
#define KSTR  68    // wt / xb row stride in floats: 68 mod 64 == 4 -> conflict-free b64 reads
#define WTSZ  (W_ * KSTR)

typedef __attribute__((ext_vector_type(2))) float v2f_t;
typedef __attribute__((ext_vector_type(8))) float v8f_t;

__device__ __forceinline__ v8f_t wmma_f32(v2f_t a, v2f_t b, v8f_t c) {
  // D = A(16x4,f32) * B(4x16,f32) + C(16x16,f32)
  return __builtin_amdgcn_wmma_f32_16x16x4_f32(
      /*neg_a=*/false, a, /*neg_b=*/false, b,
      /*c_mod=*/(short)0, c, /*reuse_a=*/false, /*reuse_b=*/false);
}

// Async copy one dword global -> LDS (scatter-capable: per-lane LDS dest).
// Tracked by ASYNCcnt; completes without touching VGPRs.
__device__ __forceinline__ void async_g2l_b32(uint32_t lds_byte_addr,
                                              uint32_t gbyte_off,
                                              const float* gbase) {
  asm volatile("global_load_async_to_lds_b32 %0, %1, %2"
               :: "v"(lds_byte_addr), "v"(gbyte_off), "s"(gbase)
               : "memory");
}

__device__ __forceinline__ void wait_async0() {
  asm volatile("s_wait_asynccnt 0x0" ::: "memory");
}

__device__ __forceinline__ uint32_t lds_addr(const float* p) {
  // flat pointer to __shared__: low 32 bits are the wave-relative LDS byte address
  return (uint32_t)(size_t)p;
}

// Stage a transposed 64-row x 256-col weight chunk: wt[c*KSTR + k] = wg[(kc*64+k)*256 + c]
__device__ __forceinline__ void issue_w_async(const float* wg, float* wtbuf,
                                              int kc, int tid) {
  for (int idx = tid; idx < 64 * W_; idx += 256) {
    int k = idx >> 8, c = idx & 255;
    async_g2l_b32(lds_addr(&wtbuf[c * KSTR + k]),
                  (uint32_t)(((kc * 64 + k) * W_ + c) * 4), wg);
  }
}

// Stage a transposed 64-row x 16-col chunk of Wlast
__device__ __forceinline__ void issue_wlast_async(const float* wg, float* wtbuf,
                                                  int kc, int tid) {
  for (int idx = tid; idx < 64 * Z_; idx += 256) {
    int k = idx >> 4, c = idx & 15;
    async_g2l_b32(lds_addr(&wtbuf[c * KSTR + k]),
                  (uint32_t)(((kc * 64 + k) * Z_ + c) * 4), wg);
  }
}

__global__ __launch_bounds__(256, 1)
void fused_grouped_mlp(const float* __restrict__ x,
                       const float* __restrict__ W0,
                       const float* __restrict__ b0,
                       const float* __restrict__ Wmid,
                       const float* __restrict__ bmid,
                       const float* __restrict__ Wlast,
                       const float* __restrict__ blast,
                       float* __restrict__ out) {
  __shared__ float h_buf[TS * HSTR];   // 133,120 B : activations, 128 tokens x 256
  __shared__ float wt2[2 * WTSZ];      // 139,264 B : double-buffered transposed K-chunks
  __shared__ float xb[TS * KSTR];      //  34,816 B : staged input tile (I padded to 64)
                                       // total 307,200 B  (<= 320KB/WG)

  const int tid  = threadIdx.x;
  const int lane = tid & 31;
  const int wave = tid >> 5;
  const int lrow = lane & 15;   // row within fragment half
  const int lhi  = lane >> 4;   // lane half -> K offset +2 / C rows +8
  const int m0   = wave * 16;   // each wave owns one exclusive 16-token m-tile

  const int sb = blockIdx.x;
  const int n  = blockIdx.y;
  const int b  = blockIdx.z;
  const int s0 = sb * TS;

  const v8f_t vzero = {0.f, 0.f, 0.f, 0.f, 0.f, 0.f, 0.f, 0.f};

  // ------------- stage 0 (async): x tile + W0^T into buf0; zero pads via ds_store -------------
  {
    const float* xg = x + (((size_t)b * N_ + n) * S_ + s0) * I_;
    for (int idx = tid; idx < TS * 64; idx += 256) {
      int tok = idx >> 6, i = idx & 63;
      if (i < I_) async_g2l_b32(lds_addr(&xb[tok * KSTR + i]),
                                (uint32_t)((tok * I_ + i) * 4), xg);
      else        xb[tok * KSTR + i] = 0.0f;
    }
    const float* wg = W0 + (size_t)n * I_ * W_;
    for (int idx = tid; idx < 64 * W_; idx += 256) {
      int k = idx >> 8, c = idx & 255;
      if (k < I_) async_g2l_b32(lds_addr(&wt2[c * KSTR + k]),
                                (uint32_t)((k * W_ + c) * 4), wg);
      else        wt2[c * KSTR + k] = 0.0f;
    }
  }
  wait_async0();
  __syncthreads();

  v8f_t acc[16];

  // ---------------- layer 0: [128 x 64pad] x [64pad x 256], compute from buf0 ----------------
  issue_w_async(Wmid + (size_t)n * (W_ * W_), &wt2[WTSZ], 0, tid);  // prefetch d=0,kc=0 -> buf1
  #pragma unroll
  for (int t = 0; t < 16; ++t) acc[t] = vzero;
  for (int kk = 0; kk < 16; ++kk) {
    const int kb = kk * 4 + lhi * 2;
    v2f_t a = *(const v2f_t*)&xb[(m0 + lrow) * KSTR + kb];
    #pragma unroll
    for (int t = 0; t < 16; ++t) {
      v2f_t bb = *(const v2f_t*)&wt2[(t * 16 + lrow) * KSTR + kb];
      acc[t] = wmma_f32(a, bb, acc[t]);
    }
  }
  {
    const float* bg = b0 + (size_t)n * W_;
    #pragma unroll
    for (int t = 0; t < 16; ++t) {
      const int col  = t * 16 + lrow;
      const float bv = bg[col];
      #pragma unroll
      for (int v = 0; v < 8; ++v) {
        const int row = m0 + v + 8 * lhi;     // own rows only: no cross-wave hazard
        h_buf[row * HSTR + col] = fmaxf(acc[t][v] + bv, 0.0f);
      }
    }
  }
  wait_async0();
  __syncthreads();

  // ------ 6 middle layers: chunk kc computed from buf[(kc+1)&1], prefetch into buf[kc&1] ------
  for (int d = 0; d < D_ - 2; ++d) {
    const float* wg = Wmid + ((size_t)d * N_ + n) * (W_ * W_);
    #pragma unroll
    for (int t = 0; t < 16; ++t) acc[t] = vzero;
    for (int kc = 0; kc < 4; ++kc) {
      float* pref = &wt2[(kc & 1) * WTSZ];
      if (kc < 3)          issue_w_async(wg, pref, kc + 1, tid);
      else if (d < D_ - 3) issue_w_async(Wmid + ((size_t)(d + 1) * N_ + n) * (W_ * W_),
                                         pref, 0, tid);
      else                 issue_wlast_async(Wlast + (size_t)n * (W_ * Z_), pref, 0, tid);
      const float* wb = &wt2[((kc + 1) & 1) * WTSZ];
      for (int kk = 0; kk < 16; ++kk) {
        const int kb = kk * 4 + lhi * 2;
        v2f_t a = *(const v2f_t*)&h_buf[(m0 + lrow) * HSTR + kc * 64 + kb];
        #pragma unroll
        for (int t = 0; t < 16; ++t) {
          v2f_t bb = *(const v2f_t*)&wb[(t * 16 + lrow) * KSTR + kb];
          acc[t] = wmma_f32(a, bb, acc[t]);
        }
      }
      wait_async0();       // own async prefetch landed
      __syncthreads();     // everyone done reading this chunk's buffer
    }
    // each wave reads/writes only its own 16 h rows; per-wave DS ordering suffices
    const float* bg = bmid + ((size_t)d * N_ + n) * W_;
    #pragma unroll
    for (int t = 0; t < 16; ++t) {
      const int col  = t * 16 + lrow;
      const float bv = bg[col];
      #pragma unroll
      for (int v = 0; v < 8; ++v) {
        const int row = m0 + v + 8 * lhi;
        h_buf[row * HSTR + col] = fmaxf(acc[t][v] + bv, 0.0f);
      }
    }
  }

  // ---------------- last layer: [128x256] x [256x16], no activation ----------------
  v8f_t accL = vzero;
  {
    const float* wg = Wlast + (size_t)n * (W_ * Z_);
    for (int kc = 0; kc < 4; ++kc) {
      if (kc < 3) issue_wlast_async(wg, &wt2[(kc & 1) * WTSZ], kc + 1, tid);
      const float* wb = &wt2[((kc + 1) & 1) * WTSZ];
      for (int kk = 0; kk < 16; ++kk) {
        const int kb = kk * 4 + lhi * 2;
        v2f_t a  = *(const v2f_t*)&h_buf[(m0 + lrow) * HSTR + kc * 64 + kb];
        v2f_t bb = *(const v2f_t*)&wb[lrow * KSTR + kb];
        accL = wmma_f32(a, bb, accL);
      }
      wait_async0();
      __syncthreads();
    }
  }
  {
    const float bv = blast[(size_t)n * Z_ + lrow];
    float* og = out + (((size_t)b * N_ + n) * S_ + s0) * Z_;
    #pragma unroll
    for (int v = 0; v < 8; ++v) {
      const int row = m0 + v + 8 * lhi;
      og[row * Z_ + lrow] = accL[v] + bv;     // 64B-contiguous per lane half
    }
  }
}

extern "C" void kernel_launch(void* const* d_in, const int* in_sizes, int n_in,
                              void* d_out, int out_size, void* d_ws, size_t ws_size,
                              hipStream_t stream) {
  (void)in_sizes; (void)n_in; (void)out_size; (void)d_ws; (void)ws_size;
  const float* x     = (const float*)d_in[0];
  const float* W0    = (const float*)d_in[1];
  const float* b0    = (const float*)d_in[2];
  const float* Wmid  = (const float*)d_in[3];
  const float* bmid  = (const float*)d_in[4];
  const float* Wlast = (const float*)d_in[5];
  const float* blast = (const float*)d_in[6];
  dim3 grid(S_ / TS, N_, B_);   // 32 x 25 x 2 = 1600 workgroups
  fused_grouped_mlp<<<grid, 256, 0, stream>>>(x, W0, b0, Wmid, bmid, Wlast, blast,
                                              (float*)d_out);
}